// SC2_KNN_6201932775870
// MI455X (gfx1250) — compile-verified
//
#include <hip/hip_runtime.h>
#include <math.h>

namespace {
constexpr int kGX  = 800;
constexpr int kGY  = 800;
constexpr int kGZ  = 45;
constexpr int kN   = 200000;
constexpr int kYZ  = kGY * kGZ;          // 36,000
constexpr int kVol = kGX * kGY * kGZ;    // 28,800,000  (fits in int32; 2*kVol too)
}

__global__ __launch_bounds__(256) void sc2_knn_gather_kernel(
    const float* __restrict__ pc1,
    const float* __restrict__ pred_flow,
    const float* __restrict__ pc2,
    const int*   __restrict__ full_ids,     // (3, GX, GY, GZ)
    const int*   __restrict__ orig_grid,    // (GX, GY, GZ)
    float*       __restrict__ out_dist,     // N floats
    int*         __restrict__ out_idx)      // N int32 (bit-exact), right after dist
{
    const int i = blockIdx.x * blockDim.x + threadIdx.x;
    if (i >= kN) return;

    // Collectively warm the whole 2.4 MB pc2 into L2 (192 MB) *before* the
    // dependent gather chain resolves: gfx1250 global_prefetch_b8, speculative,
    // no LOADcnt cost. Breaks the 4th serialized HBM round-trip.
    __builtin_prefetch(&pc2[3 * i], 0, 3);

    // Streaming, coalesced: 24 B/thread (compiler merges to b96 loads).
    const float dx = pc1[3 * i + 0] + pred_flow[3 * i + 0];
    const float dy = pc1[3 * i + 1] + pred_flow[3 * i + 1];
    const float dz = pc1[3 * i + 2] + pred_flow[3 * i + 2];

    // Voxelize: IEEE f32 divide by 0.1f (matches XLA), trunc-toward-zero
    // (matches .astype(int32)), clamp (matches jnp.clip).
    int cx = (int)((dx + 40.0f) / 0.1f);
    int cy = (int)((dy + 40.0f) / 0.1f);
    int cz = (int)((dz +  1.0f) / 0.1f);
    cx = min(max(cx, 0), kGX - 1);
    cy = min(max(cy, 0), kGY - 1);
    cz = min(max(cz, 0), kGZ - 1);

    const int cell = cx * kYZ + cy * kGZ + cz;

    // 345 MB array, random single-use gathers -> non-temporal (don't pollute L2).
    const int nx = __builtin_nontemporal_load(full_ids + cell);
    const int ny = __builtin_nontemporal_load(full_ids + kVol + cell);
    const int nz = __builtin_nontemporal_load(full_ids + 2 * kVol + cell);

    const int cell2 = nx * kYZ + ny * kGZ + nz;

    // 115 MB array, random single-use gather -> non-temporal.
    const int nn = __builtin_nontemporal_load(orig_grid + cell2);

    // pc2 gather: hits L2 thanks to the prefetch above (regular temporal).
    const float qx = pc2[3 * nn + 0] - dx;
    const float qy = pc2[3 * nn + 1] - dy;
    const float qz = pc2[3 * nn + 2] - dz;

    out_dist[i] = sqrtf(qx * qx + qy * qy + qz * qz);
    out_idx[i]  = nn;
}

extern "C" void kernel_launch(void* const* d_in, const int* in_sizes, int n_in,
                              void* d_out, int out_size, void* d_ws, size_t ws_size,
                              hipStream_t stream) {
    // setup_inputs() order: pc1, pred_flow, pc2, full_ids, orig_index_grid
    const float* pc1       = (const float*)d_in[0];
    const float* pred_flow = (const float*)d_in[1];
    const float* pc2       = (const float*)d_in[2];
    const int*   full_ids  = (const int*)d_in[3];
    const int*   orig_grid = (const int*)d_in[4];

    // Tuple output (dist f32[N], nn_indices i32[N]) concatenated flat.
    float* out_dist = (float*)d_out;
    int*   out_idx  = (int*)d_out + kN;   // int32 bits, same 4-byte slots

    const int threads = 256;                       // 8 wave32s per block
    const int blocks  = (kN + threads - 1) / threads;
    sc2_knn_gather_kernel<<<blocks, threads, 0, stream>>>(
        pc1, pred_flow, pc2, full_ids, orig_grid, out_dist, out_idx);
}